// HexCrop_50964081935402
// MI455X (gfx1250) — compile-verified
//
#include <hip/hip_runtime.h>
#include <hip/hip_bf16.h>

#define K    31
#define KK   961          // 31*31
#define HW   64
#define HWHW 4096         // 64*64
#define CC   15
#define CHB  8            // channels per block
#define TPB  256          // 8 waves (wave32)

// ---- CDNA5 async global->LDS path (probe via __has_builtin, safe fallback) ----
#if defined(__has_builtin)
#  if __has_builtin(__builtin_amdgcn_global_load_async_to_lds_b32)
#    define HAVE_ASYNC_LDS 1
#  else
#    define HAVE_ASYNC_LDS 0
#  endif
#  if __has_builtin(__builtin_amdgcn_s_wait_asynccnt)
#    define HAVE_WAIT_ASYNC 1
#  else
#    define HAVE_WAIT_ASYNC 0
#  endif
#else
#  define HAVE_ASYNC_LDS 0
#  define HAVE_WAIT_ASYNC 0
#endif

__global__ void __launch_bounds__(TPB)
hexcrop_kernel(const float* __restrict__ in,
               const int*   __restrict__ centers,
               const float* __restrict__ cmask,
               const int*   __restrict__ maskflag,
               float* __restrict__ out,
               int C)
{
    __shared__ float smask[KK];
    const int tid = threadIdx.x;
    const int b   = blockIdx.y;
    const int c0  = blockIdx.x * CHB;

    // Stage the 31x31 crop mask into LDS, reused across CHB channel planes.
    for (int i = tid; i < KK; i += TPB) {
#if HAVE_ASYNC_LDS
        __builtin_amdgcn_global_load_async_to_lds_b32(
            (__attribute__((address_space(1))) int*)(cmask + i),
            (__attribute__((address_space(3))) int*)(&smask[i]),
            /*offset=*/0, /*cpol=*/0);
#else
        smask[i] = cmask[i];
#endif
    }
#if HAVE_ASYNC_LDS
#  if HAVE_WAIT_ASYNC
    __builtin_amdgcn_s_wait_asynccnt(0);
#  else
    asm volatile("s_wait_asynccnt 0" ::: "memory");
#  endif
#endif
    __syncthreads();

    // Block-uniform scalars (compiler emits s_load for these).
    const int cy = centers[2 * b + 0];
    const int cx = centers[2 * b + 1];
    const int mf = maskflag[0];

    const long plane_in  = (long)(b * C + c0) * HWHW;
    const long plane_out = (long)(b * C + c0) * KK;
    const int  nch       = (C - c0) < CHB ? (C - c0) : CHB;
    const bool pf_ok     = (c0 + CHB) < C;

    for (int p0 = 0; p0 < KK; p0 += TPB) {
        const int p = p0 + tid;
        if (p >= KK) continue;

        const int u = p / K;
        const int v = p - u * K;
        const int y = cy + u - CC;
        const int x = cx + v - CC;
        const bool inb = ((unsigned)y < HW) && ((unsigned)x < HW);

        const float m    = mf ? smask[p] : 1.0f;
        const long  ioff = plane_in  + (long)y * HW + x;  // only deref'd when inb
        const long  ooff = plane_out + p;

        // Speculative prefetch of the next channel group's line (global_prefetch_b8).
        if (inb && pf_ok)
            __builtin_prefetch(in + ioff + (long)CHB * HWHW, 0, 1);

#pragma unroll
        for (int ch = 0; ch < CHB; ++ch) {
            if (ch < nch) {
                float val = 0.0f;
                if (inb) val = in[ioff + (long)ch * HWHW];
                out[ooff + (long)ch * KK] = val * m;
            }
        }
    }
}

// Second tuple output: the crop mask itself, appended after the cropped tensor.
__global__ void __launch_bounds__(TPB)
copymask_kernel(const float* __restrict__ cmask, float* __restrict__ out, long offset)
{
    const int i = blockIdx.x * blockDim.x + threadIdx.x;
    if (i < KK) out[offset + i] = cmask[i];
}

extern "C" void kernel_launch(void* const* d_in, const int* in_sizes, int n_in,
                              void* d_out, int out_size, void* d_ws, size_t ws_size,
                              hipStream_t stream)
{
    const float* in      = (const float*)d_in[0];  // (B,C,64,64) f32
    const int*   centers = (const int*)d_in[1];    // (B,2) i32
    const float* cmask   = (const float*)d_in[2];  // (31,31) f32
    const int*   mflag   = (const int*)d_in[3];    // scalar int
    float*       out     = (float*)d_out;

    const int B = in_sizes[1] / 2;                               // 256
    const int C = (int)(in_sizes[0] / ((long)B * HWHW));         // 256

    dim3 grid((C + CHB - 1) / CHB, B, 1);
    hexcrop_kernel<<<grid, TPB, 0, stream>>>(in, centers, cmask, mflag, out, C);

    const long tail = (long)B * C * KK;
    copymask_kernel<<<(KK + TPB - 1) / TPB, TPB, 0, stream>>>(cmask, out, tail);
}